// GCN_4303557231207
// MI455X (gfx1250) — compile-verified
//
#include <hip/hip_runtime.h>
#include <hip/hip_bf16.h>
#include <cstdint>
#include <cstddef>

// ---------------------------------------------------------------------------
// 2-layer GCN (PyG GCNConv semantics) for MI455X / gfx1250, wave32.
// GEMMs use V_WMMA_F32_16X16X4_F32 (exact fp32 path — workload is
// scatter/L2-bound, so full-precision tensor ops are the right choice).
// W panels are staged in LDS (shared by the 4 waves of a block); each wave
// computes two 16-row M tiles so every B fragment feeds two WMMAs.
// Assumes n % 16 == 0 (true for n = 50000).
// ---------------------------------------------------------------------------

typedef __attribute__((ext_vector_type(2))) float v2f;
typedef __attribute__((ext_vector_type(8))) float v8f;

// ---------------- degree / normalization ----------------

__global__ void init_deg(float* __restrict__ deg, int n) {
    int i = blockIdx.x * blockDim.x + threadIdx.x;
    if (i < n) deg[i] = 1.0f;   // self-loop contributes 1 to every node
}

__global__ void accum_deg(const long long* __restrict__ dst,
                          float* __restrict__ deg, int ne) {
    int e = blockIdx.x * blockDim.x + threadIdx.x;
    if (e < ne) atomicAdd(&deg[(int)dst[e]], 1.0f);
}

__global__ void make_dinv(const float* __restrict__ deg,
                          float* __restrict__ dinv, int n) {
    int i = blockIdx.x * blockDim.x + threadIdx.x;
    if (i < n) {
        float d = deg[i];
        dinv[i] = (d > 0.0f) ? rsqrtf(d) : 0.0f;
    }
}

// ---------------- dense GEMM via f32 WMMA + LDS-staged W ----------------
// H[n, NCOL] = X[n, K] @ W[K, NCOL], all row-major fp32.
// Block = 128 threads = 4 waves. Each wave owns two M tiles (32 rows) across
// all NCOL/16 column tiles. W is staged KC rows at a time into LDS and shared
// by all 4 waves. K-step = 4 (V_WMMA_F32_16X16X4_F32).
//
// Wave32 fragment layout (ISA 7.12.2):
//   A 16x4 : lane = half*16 + r ; a[v] = X[row0+r][k0 + 2*half + v]
//   B 4x16 : b[v] = W[k0 + 2*half + v][col0 + r]
//   D 16x16: acc[v] = H[row0 + v + 8*half][col0 + r]
template <int K, int NCOL>
__global__ __launch_bounds__(128) void gemm_wmma_f32(
    const float* __restrict__ X, const float* __restrict__ W,
    float* __restrict__ H, int n)
{
    constexpr int NT = NCOL / 16;   // column tiles
    constexpr int KC = 32;          // K rows staged per LDS chunk
    __shared__ __align__(16) float lw[KC * NCOL];   // 16KB (NCOL=128)

    const int tid  = threadIdx.x;
    const int wave = tid >> 5;
    const int lane = tid & 31;
    const int half = lane >> 4;     // 0 or 1
    const int r    = lane & 15;

    const int ntile = n >> 4;                    // n % 16 == 0 assumed
    const int m0 = blockIdx.x * 8 + wave * 2;    // this wave's M tiles
    const int m1 = m0 + 1;
    const bool v0 = (m0 < ntile);
    const bool v1 = (m1 < ntile);                // wave-uniform guards
    const int row0 = (v0 ? m0 : 0) * 16;         // clamped: loads stay in-bounds
    const int row1 = (v1 ? m1 : 0) * 16;

    const float* x0 = X + (size_t)(row0 + r) * K + 2 * half;
    const float* x1 = X + (size_t)(row1 + r) * K + 2 * half;

    v8f acc0[NT] = {};
    v8f acc1[NT] = {};

    for (int kc = 0; kc < K; kc += KC) {
        __syncthreads();            // protect lw from previous round's readers
        // Cooperative stage of W[kc .. kc+KC) x [0 .. NCOL) -> LDS (b128 ops)
        const float* wsrc = W + (size_t)kc * NCOL;
        for (int idx = tid * 4; idx < KC * NCOL; idx += 128 * 4) {
            *(float4*)&lw[idx] = *(const float4*)&wsrc[idx];
        }
        __syncthreads();

        for (int kk = 0; kk < KC; kk += 4) {
            v2f a0 = *(const v2f*)(x0 + kc + kk);   // global_load_b64
            v2f a1 = *(const v2f*)(x1 + kc + kk);
            const float* wp = &lw[(kk + 2 * half) * NCOL + r];
#pragma unroll
            for (int t = 0; t < NT; ++t) {
                v2f b;                               // paired LDS reads, stride NCOL
                b.x = wp[t * 16];
                b.y = wp[t * 16 + NCOL];
                acc0[t] = __builtin_amdgcn_wmma_f32_16x16x4_f32(
                    false, a0, false, b, (short)0, acc0[t], false, false);
                acc1[t] = __builtin_amdgcn_wmma_f32_16x16x4_f32(
                    false, a1, false, b, (short)0, acc1[t], false, false);
            }
        }
    }

    if (v0) {
#pragma unroll
        for (int t = 0; t < NT; ++t)
#pragma unroll
            for (int v = 0; v < 8; ++v)
                H[(size_t)(row0 + v + 8 * half) * NCOL + t * 16 + r] = acc0[t][v];
    }
    if (v1) {
#pragma unroll
        for (int t = 0; t < NT; ++t)
#pragma unroll
            for (int v = 0; v < 8; ++v)
                H[(size_t)(row1 + v + 8 * half) * NCOL + t * 16 + r] = acc1[t][v];
    }
}

// ---------------- edge gather/scale/scatter-add ----------------
// One thread per (edge, feature). Consecutive lanes cover consecutive
// features of the same edge -> coalesced gather of h[src] and coalesced
// float atomics into acc[dst]. acc arrays fit in the 192 MB L2.
template <int LOG_F>
__global__ void edge_scatter(const long long* __restrict__ src,
                             const long long* __restrict__ dst,
                             const float* __restrict__ dinv,
                             const float* __restrict__ h,
                             float* __restrict__ acc, int ne)
{
    constexpr int F = 1 << LOG_F;
    long long gid = (long long)blockIdx.x * blockDim.x + threadIdx.x;
    int e = (int)(gid >> LOG_F);
    int f = (int)(gid & (F - 1));
    if (e >= ne) return;
    int s = (int)src[e];
    int d = (int)dst[e];
    float w = dinv[s] * dinv[d];
    atomicAdd(&acc[(size_t)d * F + f], h[(size_t)s * F + f] * w);
}

// ---------------- layer finalizers ----------------
// Layer 1: a = relu(edge_acc + h*dinv^2 + b) * clip(p,0,1)   (in place)
template <int LOG_F>
__global__ void finalize_relu_drop(float* __restrict__ acc,
                                   const float* __restrict__ h,
                                   const float* __restrict__ dinv,
                                   const float* __restrict__ bias,
                                   const float* __restrict__ p, int n)
{
    constexpr int F = 1 << LOG_F;
    long long gid = (long long)blockIdx.x * blockDim.x + threadIdx.x;
    int i = (int)(gid >> LOG_F);
    int f = (int)(gid & (F - 1));
    if (i >= n) return;
    float di = dinv[i];
    size_t idx = (size_t)i * F + f;
    float v = acc[idx] + h[idx] * di * di + bias[f];
    v = fmaxf(v, 0.0f);
    float pf = fminf(fmaxf(p[f], 0.0f), 1.0f);
    acc[idx] = v * pf;
}

// Layer 2: out = edge_acc + g*dinv^2 + b   (in place on d_out)
template <int LOG_F>
__global__ void finalize_out(float* __restrict__ acc,
                             const float* __restrict__ g,
                             const float* __restrict__ dinv,
                             const float* __restrict__ bias, int n)
{
    constexpr int F = 1 << LOG_F;
    long long gid = (long long)blockIdx.x * blockDim.x + threadIdx.x;
    int i = (int)(gid >> LOG_F);
    int f = (int)(gid & (F - 1));
    if (i >= n) return;
    float di = dinv[i];
    size_t idx = (size_t)i * F + f;
    acc[idx] = acc[idx] + g[idx] * di * di + bias[f];
}

// ---------------- host-side orchestration ----------------

extern "C" void kernel_launch(void* const* d_in, const int* in_sizes, int n_in,
                              void* d_out, int out_size, void* d_ws, size_t ws_size,
                              hipStream_t stream)
{
    const float*     x  = (const float*)d_in[0];                 // [n, 512]
    const long long* ei = (const long long*)d_in[1];             // [2, E] int64 per reference
    const float*     W1 = (const float*)d_in[2];                 // [512, 128]
    const float*     b1 = (const float*)d_in[3];                 // [128]
    const float*     W2 = (const float*)d_in[4];                 // [128, 64]
    const float*     b2 = (const float*)d_in[5];                 // [64]
    const float*     p  = (const float*)d_in[6];                 // [128]

    const int n  = in_sizes[0] / 512;     // 50000  (multiple of 16)
    const int ne = in_sizes[1] / 2;       // 800000
    const long long* src = ei;            // edge_index[0]
    const long long* dst = ei + ne;       // edge_index[1]
    float* out = (float*)d_out;           // [n, 64]

    // Workspace layout (floats): deg | dinv | h1[n*128] | a1[n*128] | g[n*64]
    float* deg  = (float*)d_ws;
    float* dinv = deg + n;
    float* h1   = dinv + n;
    float* a1   = h1 + (size_t)n * 128;
    float* g    = a1 + (size_t)n * 128;

    // Zero the two scatter accumulators (poisoned by harness; capture-legal).
    hipMemsetAsync(a1,  0, (size_t)n * 128 * sizeof(float), stream);
    hipMemsetAsync(out, 0, (size_t)n * 64  * sizeof(float), stream);

    // Degrees + symmetric normalization (recomputed every call).
    init_deg <<<(n  + 255) / 256, 256, 0, stream>>>(deg, n);
    accum_deg<<<(ne + 255) / 256, 256, 0, stream>>>(dst, deg, ne);
    make_dinv<<<(n  + 255) / 256, 256, 0, stream>>>(deg, dinv, n);

    const int ntile  = n / 16;                 // 3125 M tiles
    const int gblk   = (ntile + 7) / 8;        // 8 M tiles per block (4 waves x 2)

    // Layer 1: h1 = x @ W1  (f32 WMMA), edge scatter, self-loop+bias+relu+drop
    gemm_wmma_f32<512, 128><<<gblk, 128, 0, stream>>>(x, W1, h1, n);
    {
        long long work = (long long)ne * 128;
        edge_scatter<7><<<(int)((work + 255) / 256), 256, 0, stream>>>(
            src, dst, dinv, h1, a1, ne);
    }
    {
        long long work = (long long)n * 128;
        finalize_relu_drop<7><<<(int)((work + 255) / 256), 256, 0, stream>>>(
            a1, h1, dinv, b1, p, n);
    }

    // Layer 2: g = a1 @ W2 (f32 WMMA), edge scatter into out, self-loop+bias
    gemm_wmma_f32<128, 64><<<gblk, 128, 0, stream>>>(a1, W2, g, n);
    {
        long long work = (long long)ne * 64;
        edge_scatter<6><<<(int)((work + 255) / 256), 256, 0, stream>>>(
            src, dst, dinv, g, out, ne);
    }
    {
        long long work = (long long)n * 64;
        finalize_out<6><<<(int)((work + 255) / 256), 256, 0, stream>>>(
            out, g, dinv, b2, n);
    }
}